// FeatureCloudEmbedder_35373350650667
// MI455X (gfx1250) — compile-verified
//
#include <hip/hip_runtime.h>
#include <math.h>

// ---------------------------------------------------------------------------
// FeatureCloudEmbedder for MI455X (gfx1250, wave32, WMMA + TDM)
//
// d2 tiles via V_WMMA_F32_16X16X4_F32:
//   A (16 candidates x K=4) row m = [cx, cy, cz, |c|^2]
//   B (K=4 x 16 queries)  col n = [-2qx, -2qy, -2qz, 1]
//   C preloaded with |q|^2  =>  D[m][n] = |c|^2 + |q|^2 - 2 c.q = d2
//
// pcd is staged into LDS in double-buffered chunks by the Tensor Data Mover
// (tensor_load_to_lds, tracked by TENSORcnt): wave 0 issues the DMA for
// chunk i+1 while all 4 waves of the block scan chunk i from LDS.
// ---------------------------------------------------------------------------

typedef __attribute__((ext_vector_type(2))) float v2f;
typedef __attribute__((ext_vector_type(8))) float v8f;
typedef __attribute__((ext_vector_type(4))) unsigned int v4u;
typedef __attribute__((ext_vector_type(8))) int v8i_t;
typedef __attribute__((ext_vector_type(4))) int v4i_t;

#define P_TOTAL 8192
#define N_PTS   16384
#define C_FEAT  64
#define K_NN    10
#define L_PE    8
#define RADIUS  0.1f
#define R2      (RADIUS * RADIUS)
#define EPSW    1e-8f
#define OUT_W   (C_FEAT + 3 + 6 * L_PE)   // 64 + 51 = 115

#define WAVES_PER_BLOCK 4
#define Q_PER_WAVE      16
#define Q_PER_BLOCK     (WAVES_PER_BLOCK * Q_PER_WAVE)

#define CHUNK   2048                       // candidates per staged LDS chunk
#define NCHUNK  (N_PTS / CHUNK)            // 8
#define CH_F    (CHUNK * 3)                // floats per chunk (6144, fits 16b)

#define HAVE_TDM __has_builtin(__builtin_amdgcn_tensor_load_to_lds)

#if HAVE_TDM
// Build a D# for a 1-row tile of CH_F consecutive f32 and issue the DMA.
// Bit layout per CDNA5 ISA 8.3/8.4 (group0: count/lds/global/type,
// group1: data_size=4B, tensor_dim0=tile_dim0=CH_F, dims1+=1/unused).
// This toolchain exposes the 6-operand builtin:
//   (v4u g0, v8i g1, v4i g2, v4i g3, v8i extra, i32 cpol)
__device__ __forceinline__ void tdm_load_chunk(const float* gsrc, unsigned lds_off)
{
    unsigned long long ga = (unsigned long long)(uintptr_t)gsrc;
    v4u g0;
    g0[0] = 1u;                                            // count=1, user desc
    g0[1] = lds_off;                                       // lds_addr (bytes)
    g0[2] = (unsigned)(ga & 0xFFFFFFFFu);                  // global_addr[31:0]
    g0[3] = (unsigned)((ga >> 32) & 0x01FFFFFFu)           // global_addr[56:32]
          | (2u << 30);                                    // type=2 ("image")
    v8i_t g1;
    g1[0] = (int)(2u << 16);                               // data_size=2 (4B)
    g1[1] = (int)((unsigned)(CH_F & 0xFFFF) << 16);        // tensor_dim0[15:0]
    g1[2] = (int)(((unsigned)CH_F >> 16) & 0xFFFFu)        // tensor_dim0[31:16]
          | (1 << 16);                                     // tensor_dim1 = 1
    g1[3] = (int)((unsigned)(CH_F & 0xFFFF) << 16);        // tile_dim0 = CH_F
    g1[4] = 1;                                             // tile_dim1=1, dim2=0
    g1[5] = CH_F;                                          // tensor_dim0_stride
    g1[6] = 0;
    g1[7] = 0;
    v4i_t z4 = {0, 0, 0, 0};                               // dims 2-4 unused
    v8i_t z8 = {0, 0, 0, 0, 0, 0, 0, 0};
    __builtin_amdgcn_tensor_load_to_lds(g0, g1, z4, z4, z8, 0);
}
#endif

__global__ __launch_bounds__(WAVES_PER_BLOCK * 32)
void knn_agg_kernel(const float* __restrict__ xyz,
                    const float* __restrict__ pcd,
                    const float* __restrict__ feat,
                    float* __restrict__ out)
{
    __shared__ float sPcd[2][CH_F];                         // 48 KB staging
    __shared__ float sD[WAVES_PER_BLOCK][Q_PER_WAVE][2][K_NN];
    __shared__ int   sI[WAVES_PER_BLOCK][Q_PER_WAVE][2][K_NN];
    __shared__ float sW[WAVES_PER_BLOCK][Q_PER_WAVE][K_NN];
    __shared__ int   sJ[WAVES_PER_BLOCK][Q_PER_WAVE][K_NN];

    const int tid   = threadIdx.x;
    const int wv    = tid >> 5;
    const int lane  = tid & 31;
    const int hi    = lane >> 4;        // 0 = K0/K1 + rows 0-7, 1 = K2/K3 + rows 8-15
    const int ql    = lane & 15;        // query column within the 16-query tile
    const int qbase = (blockIdx.x * WAVES_PER_BLOCK + wv) * Q_PER_WAVE;
    const int q     = qbase + ql;

    // ---- B operand (queries), constant across the whole candidate sweep ----
    const float qx = xyz[q * 3 + 0];
    const float qy = xyz[q * 3 + 1];
    const float qz = xyz[q * 3 + 2];
    const float q2 = qx * qx + qy * qy + qz * qz;
    v2f bmat;
    bmat.x = hi ? (-2.0f * qz) : (-2.0f * qx);
    bmat.y = hi ? 1.0f         : (-2.0f * qy);
    v8f cin;
#pragma unroll
    for (int r = 0; r < 8; ++r) cin[r] = q2;   // bake |q|^2 into the accumulator

    // ---- per-lane top-K (sorted ascending); sentinel = R^2 prunes ----
    float dtop[K_NN];
    int   itop[K_NN];
#pragma unroll
    for (int k = 0; k < K_NN; ++k) { dtop[k] = R2; itop[k] = 0; }

    // ---- stage chunk 0 ----
#if HAVE_TDM
    if (wv == 0) {
        tdm_load_chunk(pcd, (unsigned)(uintptr_t)&sPcd[0][0]);
        __builtin_amdgcn_s_wait_tensorcnt(0);
    }
    __syncthreads();
#else
    for (int i = tid; i < CH_F; i += WAVES_PER_BLOCK * 32) sPcd[0][i] = pcd[i];
    __syncthreads();
#endif

    for (int ch = 0; ch < NCHUNK; ++ch) {
        const int cur = ch & 1;
#if HAVE_TDM
        if (wv == 0 && ch + 1 < NCHUNK)                     // async prefetch
            tdm_load_chunk(pcd + (size_t)(ch + 1) * CH_F,
                           (unsigned)(uintptr_t)&sPcd[cur ^ 1][0]);
#endif
        const float* __restrict__ buf = sPcd[cur];
        const int gbase = ch * CHUNK;

#pragma unroll 2
        for (int n0 = 0; n0 < CHUNK; n0 += 16) {
            const int   ri = (n0 + ql) * 3;                 // lane pair, same row
            const float cx = buf[ri + 0];
            const float cy = buf[ri + 1];
            const float cz = buf[ri + 2];
            const float c2 = cx * cx + cy * cy + cz * cz;
            v2f amat;
            amat.x = hi ? cz : cx;
            amat.y = hi ? c2 : cy;

            // D = A*B + C : 16x16 tile of squared distances in one matrix op
            v8f d = __builtin_amdgcn_wmma_f32_16x16x4_f32(
                /*neg_a=*/false, amat, /*neg_b=*/false, bmat,
                /*c_mod=*/(short)0, cin, /*reuse_a=*/false, /*reuse_b=*/false);

            // one branch per 16-candidate tile; hits are rare (~6%/lane/tile)
            const float gate = dtop[K_NN - 1];
            bool hit = false;
#pragma unroll
            for (int r = 0; r < 8; ++r) hit |= (d[r] < gate);

            if (hit) {
#pragma unroll
                for (int r = 0; r < 8; ++r) {
                    const float d2v = d[r];
                    if (d2v < dtop[K_NN - 1]) {
                        float dd = d2v;
                        int   ii = gbase + n0 + r + (hi << 3);
#pragma unroll
                        for (int s = 0; s < K_NN; ++s) {    // sorted insert
                            const bool  sm = dd < dtop[s];
                            const float td = dtop[s];
                            const int   ti = itop[s];
                            dtop[s] = sm ? dd : td;
                            itop[s] = sm ? ii : ti;
                            dd      = sm ? td : dd;
                            ii      = sm ? ti : ii;
                        }
                    }
                }
            }
        }

#if HAVE_TDM
        if (wv == 0 && ch + 1 < NCHUNK)
            __builtin_amdgcn_s_wait_tensorcnt(0);           // chunk ch+1 landed
        __syncthreads();
#else
        __syncthreads();
        if (ch + 1 < NCHUNK) {
            const float* src = pcd + (size_t)(ch + 1) * CH_F;
            for (int i = tid; i < CH_F; i += WAVES_PER_BLOCK * 32)
                sPcd[cur ^ 1][i] = src[i];
            __syncthreads();
        }
#endif
    }

#pragma unroll
    for (int k = 0; k < K_NN; ++k) {
        sD[wv][ql][hi][k] = dtop[k];
        sI[wv][ql][hi][k] = itop[k];
    }
    __syncthreads();

    // ---- merge the lane pair's two sorted 10-lists, compute weights ----
    if (lane < 16) {
        int   i = 0, j = 0;
        float md[K_NN];
        int   mi[K_NN];
#pragma unroll
        for (int k = 0; k < K_NN; ++k) {
            const float da = sD[wv][lane][0][i];
            const float db = sD[wv][lane][1][j];
            const bool  ta = (da <= db);
            md[k] = ta ? da : db;
            mi[k] = ta ? sI[wv][lane][0][i] : sI[wv][lane][1][j];
            i += ta ? 1 : 0;
            j += ta ? 0 : 1;
        }
        float w[K_NN];
        float sum = 0.0f;
#pragma unroll
        for (int k = 0; k < K_NN; ++k) {
            const float dist = sqrtf(fmaxf(md[k], 0.0f) + 1e-12f);
            const float wk   = (dist < RADIUS) ? (1.0f / (dist + EPSW)) : 0.0f;
            w[k] = wk;
            sum += wk;
        }
        const float inv = 1.0f / (sum + EPSW);
#pragma unroll
        for (int k = 0; k < K_NN; ++k) {
            sW[wv][lane][k] = w[k] * inv;
            sJ[wv][lane][k] = mi[k];
        }
    }
    __syncthreads();

    // ---- weighted feature aggregation: 64 channels, 2 per lane ----
    for (int qq = 0; qq < Q_PER_WAVE; ++qq) {
        float a0 = 0.0f, a1 = 0.0f;
#pragma unroll
        for (int k = 0; k < K_NN; ++k) {
            const float  wk  = sW[wv][qq][k];          // LDS broadcast
            const int    ik  = sJ[wv][qq][k];
            const float* row = feat + (size_t)ik * C_FEAT;
            a0 += wk * row[lane];                      // coalesced 32-lane read
            a1 += wk * row[lane + 32];
        }
        float* orow = out + (size_t)(qbase + qq) * OUT_W;
        orow[lane]      = a0;
        orow[lane + 32] = a1;
    }
}

// ---- NeRF positional encoding branch: out[p][64..114] ----
__global__ __launch_bounds__(256)
void pe_kernel(const float* __restrict__ xyz, float* __restrict__ out)
{
    const int p = blockIdx.x * blockDim.x + threadIdx.x;
    if (p >= P_TOTAL) return;
    const float v0 = xyz[p * 3 + 0];
    const float v1 = xyz[p * 3 + 1];
    const float v2 = xyz[p * 3 + 2];
    float* o = out + (size_t)p * OUT_W + C_FEAT;
    o[0] = v0; o[1] = v1; o[2] = v2;
    const float vv[3] = { v0, v1, v2 };
    float f = 1.0f;
#pragma unroll
    for (int i = 0; i < L_PE; ++i) {
#pragma unroll
        for (int d = 0; d < 3; ++d) {
            float s, c;
            __sincosf(f * vv[d], &s, &c);
            o[3 + i * 3 + d]            = s;   // sin block (i,d) order
            o[3 + 3 * L_PE + i * 3 + d] = c;   // cos block
        }
        f *= 2.0f;
    }
}

extern "C" void kernel_launch(void* const* d_in, const int* in_sizes, int n_in,
                              void* d_out, int out_size, void* d_ws, size_t ws_size,
                              hipStream_t stream)
{
    const float* xyz  = (const float*)d_in[0];   // [1,8192,3]
    const float* pcd  = (const float*)d_in[1];   // [1,16384,3]
    const float* feat = (const float*)d_in[2];   // [1,16384,64]
    float*       out  = (float*)d_out;           // [1,8192,115]

    knn_agg_kernel<<<P_TOTAL / Q_PER_BLOCK, WAVES_PER_BLOCK * 32, 0, stream>>>(
        xyz, pcd, feat, out);
    pe_kernel<<<(P_TOTAL + 255) / 256, 256, 0, stream>>>(xyz, out);
}